// Channel_Attention_68307159875991
// MI455X (gfx1250) — compile-verified
//
#include <hip/hip_runtime.h>
#include <hip/hip_bf16.h>
#include <math.h>

typedef __attribute__((ext_vector_type(2))) float v2f;
typedef __attribute__((ext_vector_type(8))) float v8f;

#define BB 2
#define CC 128
#define TT 512
#define FF 512
#define HH 8
#define DD 64              // head dim of q/k
#define NPH (TT * DD)      // N per (b,h) = 32768
#define CSTRIDE ((size_t)TT * FF)   // x channel stride in floats

// ---------------- Stage 1: mean over T ----------------
__global__ void mean_kernel(const float* __restrict__ x, float* __restrict__ xm) {
    const int bc  = blockIdx.x;        // 0..B*C-1
    const int tid = threadIdx.x;       // 256 threads
    const float* xp = x + (size_t)bc * CSTRIDE;
    float a0 = 0.f, a1 = 0.f;
    for (int t = 0; t < TT; ++t) {
        const float* row = xp + (size_t)t * FF;
        a0 += row[tid];
        a1 += row[tid + 256];
    }
    const float inv = 1.0f / (float)TT;
    xm[(size_t)bc * FF + tid]       = a0 * inv;
    xm[(size_t)bc * FF + tid + 256] = a1 * inv;
}

// ---------------- Stage 2: q = xm Wq^T + bq ; k = xm Wk^T ----------------
__global__ void proj_kernel(const float* __restrict__ xm,
                            const float* __restrict__ Wq, const float* __restrict__ bq,
                            const float* __restrict__ Wk,
                            float* __restrict__ q, float* __restrict__ k) {
    const int i = blockIdx.y;                          // 0..B*C-1
    const int j = blockIdx.x * 256 + threadIdx.x;      // 0..F-1
    __shared__ float xrow[FF];
    for (int f = threadIdx.x; f < FF; f += 256) xrow[f] = xm[(size_t)i * FF + f];
    __syncthreads();
    const float* wqr = Wq + (size_t)j * FF;
    const float* wkr = Wk + (size_t)j * FF;
    float aq = 0.f, ak = 0.f;
    for (int f = 0; f < FF; ++f) {
        const float xv = xrow[f];
        aq = fmaf(xv, wqr[f], aq);
        ak = fmaf(xv, wkr[f], ak);
    }
    q[(size_t)i * FF + j] = aq + bq[j];
    k[(size_t)i * FF + j] = ak;
}

// ---------------- Stage 3: qk + softmax -> w (B,H,C,C) ----------------
__global__ void attn_kernel(const float* __restrict__ q, const float* __restrict__ k,
                            float* __restrict__ w) {
    const int bh = blockIdx.x;             // 0..B*H-1
    const int b  = bh / HH;
    const int h  = bh % HH;
    const int tid = threadIdx.x;           // 256 threads
    __shared__ float pS[CC * CC];          // 64 KB

    const float* qb = q + (size_t)b * CC * FF + h * DD;   // row c at qb + c*FF
    const float* kb = k + (size_t)b * CC * FF + h * DD;

    for (int e = tid; e < CC * CC; e += 256) {
        const int c  = e >> 7;
        const int cp = e & (CC - 1);
        const float* qr = qb + (size_t)c  * FF;
        const float* kr = kb + (size_t)cp * FF;
        float acc = 0.f;
        for (int d = 0; d < DD; ++d) acc = fmaf(qr[d], kr[d], acc);
        pS[e] = acc * 0.125f;              // d^{-1/4} applied to q and k => d^{-1/2} = 1/8
    }
    __syncthreads();

    if (tid < CC) {                        // one thread per softmax row
        float* row = pS + tid * CC;
        float m = row[0];
        for (int j = 1; j < CC; ++j) m = fmaxf(m, row[j]);
        float s = 0.f;
        for (int j = 0; j < CC; ++j) { float e = expf(row[j] - m); row[j] = e; s += e; }
        const float inv = 1.0f / s;
        for (int j = 0; j < CC; ++j) row[j] *= inv;
    }
    __syncthreads();

    float* wout = w + (size_t)bh * CC * CC;
    for (int e = tid; e < CC * CC; e += 256) wout[e] = pS[e];
}

// ---------------- Stage 4: out[b,c,t,f] = sum_c' w[b,f/64,c,c'] * x[b,c',t,f] ----------------
// Batched GEMM per (b,h): M=128, K=128, N=32768 using V_WMMA_F32_16X16X4_F32.
// Block = 256 threads = 8 waves; each wave owns a 128(M) x 16(N) output strip.
__global__ void wv_kernel(const float* __restrict__ x, const float* __restrict__ w,
                          float* __restrict__ out) {
    const int bh    = blockIdx.y;          // 0..15
    const int b     = bh / HH;
    const int h     = bh % HH;
    const int nBase = blockIdx.x * 128;    // 256 blocks cover N=32768
    const int tid   = threadIdx.x;
    const int wave  = tid >> 5;
    const int lane  = tid & 31;

    __shared__ float wS[CC * CC];          // 64 KB: w[b,h] tile, row-major (c, c')
    const float* wG = w + (size_t)bh * CC * CC;
    for (int i = tid; i < CC * CC; i += 256) wS[i] = wG[i];
    __syncthreads();

    const int laneN  = lane & 15;          // N within 16-wide tile / M within A frag
    const int laneHi = lane >> 4;          // 0 or 1 (K sub-pair select)
    const int n = nBase + wave * 16 + laneN;
    const int t = n >> 6;
    const int f = (h << 6) + (n & 63);

    // column of B-operand for this lane: x[b, c', t, f], c' varies with stride CSTRIDE
    const float* xcol = x + (size_t)b * CC * CSTRIDE + (size_t)t * FF + f;

    v8f acc[8] = {v8f{}, v8f{}, v8f{}, v8f{}, v8f{}, v8f{}, v8f{}, v8f{}};

    for (int k = 0; k < CC; k += 4) {
        const int kk = k + 2 * laneHi;     // B frag: VGPR0 holds K=kk, VGPR1 holds K=kk+1
        v2f bfrag;
        bfrag.x = xcol[(size_t)kk * CSTRIDE];
        bfrag.y = xcol[(size_t)(kk + 1) * CSTRIDE];
#pragma unroll
        for (int mt = 0; mt < 8; ++mt) {
            const int m = mt * 16 + laneN; // A frag: lane -> row M, VGPR v -> K = kk+v
            v2f a = *(const v2f*)&wS[m * CC + kk];
            acc[mt] = __builtin_amdgcn_wmma_f32_16x16x4_f32(
                false, a, false, bfrag, (short)0, acc[mt], false, false);
        }
    }

    // C/D layout: VGPR r, lane l -> M = r + 8*(l>>4), N = l&15
    float* obase = out + (size_t)b * CC * CSTRIDE + (size_t)t * FF + f;
#pragma unroll
    for (int mt = 0; mt < 8; ++mt) {
#pragma unroll
        for (int r = 0; r < 8; ++r) {
            const int c = mt * 16 + r + 8 * laneHi;
            obase[(size_t)c * CSTRIDE] = acc[mt][r];
        }
    }
}

extern "C" void kernel_launch(void* const* d_in, const int* in_sizes, int n_in,
                              void* d_out, int out_size, void* d_ws, size_t ws_size,
                              hipStream_t stream) {
    const float* x  = (const float*)d_in[0];
    const float* Wq = (const float*)d_in[1];
    const float* bq = (const float*)d_in[2];
    const float* Wk = (const float*)d_in[3];
    float* out = (float*)d_out;

    float* xm = (float*)d_ws;                        // B*C*F      = 131072 floats
    float* q  = xm + (size_t)BB * CC * FF;           // 131072 floats
    float* k  = q  + (size_t)BB * CC * FF;           // 131072 floats
    float* w  = k  + (size_t)BB * CC * FF;           // B*H*C*C    = 262144 floats
    (void)in_sizes; (void)n_in; (void)out_size; (void)ws_size;

    mean_kernel<<<dim3(BB * CC), dim3(256), 0, stream>>>(x, xm);
    proj_kernel<<<dim3(FF / 256, BB * CC), dim3(256), 0, stream>>>(xm, Wq, bq, Wk, q, k);
    attn_kernel<<<dim3(BB * HH), dim3(256), 0, stream>>>(q, k, w);
    wv_kernel<<<dim3(NPH / 128, BB * HH), dim3(256), 0, stream>>>(x, w, out);
}